// LuongAttention_32839319945312
// MI455X (gfx1250) — compile-verified
//
#include <hip/hip_runtime.h>
#include <math.h>

// ---------------- problem constants ----------------
#define B_   64
#define S_   2048
#define H_   2048
#define G_   8                 // S splits per batch
#define SCHUNK (S_ / G_)       // 256 rows per workgroup
#define TILE 16                // rows per staged tile (matches WMMA M)
#define NTILES (SCHUNK / TILE) // 16
#define PAD  4                 // LDS row padding (floats) to dodge bank conflicts
#define ROWST (H_ + PAD)       // 2052 floats per LDS tile row
#define NTHREADS 256
#define NWAVES   (NTHREADS / 32)
#define ASYNC_PER_WAVE ((TILE * H_) / (NTHREADS * 4))   // 32 b128 issues per wave per tile

typedef float v2f __attribute__((ext_vector_type(2)));
typedef float v8f __attribute__((ext_vector_type(8)));

// Types matching the async-to-LDS builtin signature (v4i pointers in AS1/AS3).
typedef int v4i_ __attribute__((vector_size(16)));
typedef v4i_ __attribute__((address_space(1)))* g128_t;   // global
typedef v4i_ __attribute__((address_space(3)))* l128_t;   // LDS

// LDS layout (floats): q[H] | tile0[TILE][ROWST] | tile1[TILE][ROWST] | pscore | score16
#define LDS_FLOATS (H_ + 2 * TILE * ROWST + NWAVES * TILE + TILE)

// ---------------------------------------------------------------------
// CDNA5 async global->LDS copy (ASYNCcnt path). Builtin when declared,
// inline asm otherwise (portable across ROCm 7.2 / amdgpu-toolchain).
// ---------------------------------------------------------------------
__device__ __forceinline__ void async_load_b128(const float* gsrc, float* ldst)
{
#if __has_builtin(__builtin_amdgcn_global_load_async_to_lds_b128)
    __builtin_amdgcn_global_load_async_to_lds_b128(
        (g128_t)(v4i_*)(void*)gsrc,
        (l128_t)(v4i_*)(void*)ldst,
        0, 0);
#else
    unsigned lds_off = (unsigned)(unsigned long long)ldst;
    unsigned long long ga = (unsigned long long)gsrc;
    asm volatile("global_load_async_to_lds_b128 %0, %1, off"
                 :: "v"(lds_off), "v"(ga) : "memory");
#endif
}

__device__ __forceinline__ void wait_async_32()
{
#if __has_builtin(__builtin_amdgcn_s_wait_asynccnt)
    __builtin_amdgcn_s_wait_asynccnt(32);
#else
    asm volatile("s_wait_asynccnt 32" ::: "memory");
#endif
}

__device__ __forceinline__ void wait_async_0()
{
#if __has_builtin(__builtin_amdgcn_s_wait_asynccnt)
    __builtin_amdgcn_s_wait_asynccnt(0);
#else
    asm volatile("s_wait_asynccnt 0" ::: "memory");
#endif
}

// Issue one tile's worth of async b128 copies (32 per wave, fully coalesced).
__device__ __forceinline__ void stage_tile_async(const float* __restrict__ src,
                                                 float* __restrict__ dst, int t)
{
#pragma unroll
    for (int k = 0; k < ASYNC_PER_WAVE; ++k) {
        const int idx = t * 4 + k * NTHREADS * 4;
        const int r = idx >> 11;          // / H_
        const int c = idx & (H_ - 1);
        async_load_b128(src + idx, dst + r * ROWST + c);
    }
}

// =====================================================================
// Kernel 1: flash-style online softmax + context accumulation with
// double-buffered async global->LDS staging.
// grid = (G_, B_), block = 256 threads (8 waves).
// Scores via V_WMMA_F32_16X16X4_F32:
//   A[m,k] = E[s0+m, h0+k]  (16x4 fp32 tile, ISA A-layout)
//   B[k,n] = q[h0+k] for all n (broadcast -> every D column = partial score)
// =====================================================================
__global__ __launch_bounds__(NTHREADS)
void attn_main_kernel(const float* __restrict__ hidden,
                      const float* __restrict__ enc,
                      float* __restrict__ ctx_part,   // [B*G][H]
                      float* __restrict__ m_part,     // [B*G]
                      float* __restrict__ l_part,     // [B*G]
                      float* __restrict__ attn_raw)   // [B][S] (d_out attn region)
{
    extern __shared__ float smem[];
    float* q_lds   = smem;                            // H_
    float* tile0   = q_lds + H_;                      // TILE * ROWST
    float* tile1   = tile0 + TILE * ROWST;            // TILE * ROWST
    float* pscore  = tile1 + TILE * ROWST;            // NWAVES * TILE
    float* score16 = pscore + NWAVES * TILE;          // TILE

    const int b    = blockIdx.y;
    const int g    = blockIdx.x;
    const int t    = threadIdx.x;
    const int wave = t >> 5;
    const int lane = t & 31;

    const float* qg = hidden + (size_t)b * H_;
    const float* eb = enc + (size_t)b * S_ * H_ + (size_t)g * SCHUNK * H_;

    // Kick off async staging of tile 0 immediately.
    stage_tile_async(eb, tile0, t);

    // Load q row into LDS (regular coalesced loads, once per workgroup).
    for (int i = t * 4; i < H_; i += NTHREADS * 4) {
        float4 v = *(const float4*)(qg + i);
        *(float4*)(q_lds + i) = v;
    }

    float m_run = -INFINITY;
    float l_run = 0.0f;
    float ctx[8];
#pragma unroll
    for (int j = 0; j < 8; ++j) ctx[j] = 0.0f;

    // A-fragment addressing (fixed per lane): row = lane%16, K-pair = lane/16.
    const int  arow_off = (lane & 15) * ROWST + ((lane >> 4) << 1);
    const int  koff     = (lane >> 4) << 1;           // 0 or 2
    const float* brow   = q_lds + koff;

    for (int tb = 0; tb < NTILES; ++tb) {
        float* cur = (tb & 1) ? tile1 : tile0;
        float* nxt = (tb & 1) ? tile0 : tile1;

        // Overlap: issue next tile's async copies, then wait only for the
        // current tile (async loads complete in order; 32 issues/wave/tile).
        if (tb + 1 < NTILES) {
            stage_tile_async(eb + (size_t)(tb + 1) * TILE * H_, nxt, t);
            wait_async_32();
        } else {
            wait_async_0();
        }
        __syncthreads();   // cross-wave visibility of cur; guards nxt reuse

        // ---- Phase A: WMMA partial scores; wave w owns h in [w*256, w*256+256) ----
        v8f acc = {0.f, 0.f, 0.f, 0.f, 0.f, 0.f, 0.f, 0.f};
        const float* arow = cur + arow_off;
        const int hbase = wave * 256;
#pragma unroll 8
        for (int i = 0; i < 64; ++i) {
            const int h0 = hbase + 4 * i;
            v2f a  = *(const v2f*)(arow + h0);   // E[row, h0+koff .. +1]
            v2f bq = *(const v2f*)(brow + h0);   // q[h0+koff .. +1] (bcast over N)
            acc = __builtin_amdgcn_wmma_f32_16x16x4_f32(
                /*neg_a=*/false, a, /*neg_b=*/false, bq,
                /*c_mod=*/(short)0, acc, /*reuse_a=*/false, /*reuse_b=*/false);
        }
        // D layout: VGPR r -> (M=r, N=lane) lanes 0-15 ; (M=r+8, N=lane-16) lanes 16-31.
        // All N columns are identical (B was broadcast), so lane 0 / lane 16 suffice.
        if (lane == 0) {
#pragma unroll
            for (int r = 0; r < 8; ++r) pscore[wave * TILE + r] = acc[r];
        }
        if (lane == 16) {
#pragma unroll
            for (int r = 0; r < 8; ++r) pscore[wave * TILE + 8 + r] = acc[r];
        }
        __syncthreads();

        // Cross-wave reduce -> 16 scores; stash raw scores for final attn pass.
        if (t < TILE) {
            float s = 0.0f;
#pragma unroll
            for (int w = 0; w < NWAVES; ++w) s += pscore[w * TILE + t];
            score16[t] = s;
            attn_raw[(size_t)b * S_ + (size_t)g * SCHUNK + tb * TILE + t] = s;
        }
        __syncthreads();

        // ---- Online softmax update (computed redundantly by every thread) ----
        float tmax = -INFINITY;
#pragma unroll
        for (int r = 0; r < TILE; ++r) tmax = fmaxf(tmax, score16[r]);
        const float m_new = fmaxf(m_run, tmax);
        const float scale = __expf(m_run - m_new);   // 0 on first tile (m_run=-inf)
        float wv[TILE];
        float ladd = 0.0f;
#pragma unroll
        for (int r = 0; r < TILE; ++r) {
            wv[r] = __expf(score16[r] - m_new);
            ladd += wv[r];
        }
        l_run = l_run * scale + ladd;
        m_run = m_new;

        // ---- Phase B: context accumulate from the SAME LDS tile (no re-read) ----
#pragma unroll
        for (int j = 0; j < 8; ++j) ctx[j] *= scale;
        for (int r = 0; r < TILE; ++r) {
            const float w = wv[r];
            const float* trow = cur + r * ROWST + t;
#pragma unroll
            for (int j = 0; j < 8; ++j)
                ctx[j] = fmaf(w, trow[256 * j], ctx[j]);
        }
        __syncthreads();   // all waves done reading cur -> next iter may overwrite
    }

    // ---- write per-split partials ----
    const int pg = b * G_ + g;
    if (t == 0) { m_part[pg] = m_run; l_part[pg] = l_run; }
    float* cp = ctx_part + (size_t)pg * H_;
#pragma unroll
    for (int j = 0; j < 8; ++j) cp[t + 256 * j] = ctx[j];
}

// =====================================================================
// Kernel 2: combine the G_ partials per batch; normalize attn weights.
// grid = B_, block = 256.
// =====================================================================
__global__ __launch_bounds__(256)
void attn_combine_kernel(const float* __restrict__ ctx_part,
                         const float* __restrict__ m_part,
                         const float* __restrict__ l_part,
                         float* __restrict__ ctx_out,
                         float* __restrict__ attn_out)
{
    const int b = blockIdx.x;
    const int t = threadIdx.x;

    float m_fin = -INFINITY;
#pragma unroll
    for (int g = 0; g < G_; ++g) m_fin = fmaxf(m_fin, m_part[b * G_ + g]);

    float eg[G_];
    float l_fin = 0.0f;
#pragma unroll
    for (int g = 0; g < G_; ++g) {
        eg[g] = __expf(m_part[b * G_ + g] - m_fin);
        l_fin += l_part[b * G_ + g] * eg[g];
    }
    const float inv = 1.0f / l_fin;

#pragma unroll
    for (int j = 0; j < 8; ++j) {
        const int h = t + 256 * j;
        float a = 0.0f;
#pragma unroll
        for (int g = 0; g < G_; ++g)
            a = fmaf(eg[g], ctx_part[(size_t)(b * G_ + g) * H_ + h], a);
        ctx_out[(size_t)b * H_ + h] = a * inv;
    }

    for (int s = t; s < S_; s += 256) {
        const float raw = attn_out[(size_t)b * S_ + s];
        attn_out[(size_t)b * S_ + s] = __expf(raw - m_fin) * inv;
    }
}

// =====================================================================
extern "C" void kernel_launch(void* const* d_in, const int* in_sizes, int n_in,
                              void* d_out, int out_size, void* d_ws, size_t ws_size,
                              hipStream_t stream)
{
    const float* hidden = (const float*)d_in[0];   // [1, B, H] fp32
    const float* enc    = (const float*)d_in[1];   // [B, S, H] fp32

    float* ctx_out  = (float*)d_out;               // [B, H]
    float* attn_out = ctx_out + (size_t)B_ * H_;   // [B, S]

    // workspace: m[B*G] | l[B*G] | ctx_part[B*G][H]  (~4.2 MB)
    float* m_part   = (float*)d_ws;
    float* l_part   = m_part + B_ * G_;
    float* ctx_part = l_part + B_ * G_;

    const size_t lds_bytes = (size_t)LDS_FLOATS * sizeof(float);

    dim3 grid(G_, B_);
    attn_main_kernel<<<grid, NTHREADS, lds_bytes, stream>>>(
        hidden, enc, ctx_part, m_part, l_part, attn_out);

    attn_combine_kernel<<<B_, 256, 0, stream>>>(
        ctx_part, m_part, l_part, ctx_out, attn_out);
}